// Cell_76476187673144
// MI455X (gfx1250) — compile-verified
//
#include <hip/hip_runtime.h>
#include <cstdint>
#include <cstddef>

#define BN_EPS 1e-5f
constexpr int D = 64;
constexpr int WAVES = 4;           // waves per block (wave32)
constexpr int TPB = 32 * WAVES;

typedef __attribute__((ext_vector_type(2))) float v2f;
typedef __attribute__((ext_vector_type(8))) float v8f;

__device__ __forceinline__ float leaky_f(float x) { return x > 0.f ? x : 0.2f * x; }

// ---------------------------------------------------------------- utilities
__global__ void fill_f32(float* __restrict__ p, size_t n, float v) {
    size_t i = (size_t)blockIdx.x * blockDim.x + threadIdx.x;
    size_t st = (size_t)gridDim.x * blockDim.x;
    for (; i < n; i += st) p[i] = v;
}

__global__ void softmax8x3(const float* __restrict__ w, float* __restrict__ P) {
    int r = threadIdx.x;
    if (r < 8) {
        float a = w[r * 3 + 0], b = w[r * 3 + 1], c = w[r * 3 + 2];
        float m = fmaxf(a, fmaxf(b, c));
        float ea = expf(a - m), eb = expf(b - m), ec = expf(c - m);
        float s = ea + eb + ec;
        P[r * 3 + 0] = ea / s;
        P[r * 3 + 1] = eb / s;
        P[r * 3 + 2] = ec / s;
    }
}

__global__ void fix_nonfinite(float* __restrict__ p, size_t n) {
    size_t i = (size_t)blockIdx.x * blockDim.x + threadIdx.x;
    size_t st = (size_t)gridDim.x * blockDim.x;
    for (; i < n; i += st) {
        float v = p[i];
        if (!__builtin_isfinite(v)) p[i] = 0.f;
    }
}

// float atomic max via sign-split int/uint monotone trick (init must be -inf)
__device__ __forceinline__ void atomicMaxF(float* addr, float v) {
    if (v >= 0.f) atomicMax((int*)addr, __float_as_int(v));
    else          atomicMin((unsigned int*)addr, __float_as_uint(v));
}

// ------------------------------------------------- edge message + seg sum/max
__global__ void edge_msg_agg(const float* __restrict__ V, const float* __restrict__ E,
                             const int* __restrict__ src, const int* __restrict__ dst,
                             float* __restrict__ agg_sum, float* __restrict__ agg_max, int M) {
    size_t t = (size_t)blockIdx.x * blockDim.x + threadIdx.x;
    size_t total = (size_t)M * 16;           // one thread per (edge, 4-float group)
    if (t >= total) return;
    int e = (int)(t >> 4);
    int q = ((int)t & 15) * 4;
    int s = src[e], d = dst[e];
    const float4 ev = *(const float4*)(E + (size_t)e * D + q);
    const float4 vv = *(const float4*)(V + (size_t)s * D + q);
    float m0 = vv.x * (1.f / (1.f + expf(-ev.x)));
    float m1 = vv.y * (1.f / (1.f + expf(-ev.y)));
    float m2 = vv.z * (1.f / (1.f + expf(-ev.z)));
    float m3 = vv.w * (1.f / (1.f + expf(-ev.w)));
    float* ps = agg_sum + (size_t)d * D + q;
    float* pm = agg_max + (size_t)d * D + q;
    atomicAdd(ps + 0, m0); atomicAdd(ps + 1, m1);
    atomicAdd(ps + 2, m2); atomicAdd(ps + 3, m3);
    atomicMaxF(pm + 0, m0); atomicMaxF(pm + 1, m1);
    atomicMaxF(pm + 2, m2); atomicMaxF(pm + 3, m3);
}

// ----------------------------------------------- BN finalize / apply kernels
// out += p * (Y - mean) * rsqrt(var + eps), stats = [colsum | colsumsq]
__global__ void bn_accum(const float* __restrict__ Y, const float* __restrict__ stats,
                         const float* __restrict__ P, int pidx,
                         float* __restrict__ out, int R, int NC) {
    size_t n = (size_t)R * NC;
    float p = P[pidx];
    float invR = 1.f / (float)R;
    size_t i = (size_t)blockIdx.x * blockDim.x + threadIdx.x;
    size_t st = (size_t)gridDim.x * blockDim.x;
    for (; i < n; i += st) {
        int c = (int)(i % NC);
        float mean = stats[c] * invR;
        float var = stats[NC + c] * invR - mean * mean;
        out[i] += p * (Y[i] - mean) * rsqrtf(var + BN_EPS);
    }
}

// Y = leaky((Y - mean) * rsqrt(var + eps)) in place
__global__ void bn_leaky_inplace(float* __restrict__ Y, const float* __restrict__ stats,
                                 int R, int NC) {
    size_t n = (size_t)R * NC;
    float invR = 1.f / (float)R;
    size_t i = (size_t)blockIdx.x * blockDim.x + threadIdx.x;
    size_t st = (size_t)gridDim.x * blockDim.x;
    for (; i < n; i += st) {
        int c = (int)(i % NC);
        float mean = stats[c] * invR;
        float var = stats[NC + c] * invR - mean * mean;
        float v = (Y[i] - mean) * rsqrtf(var + BN_EPS);
        Y[i] = leaky_f(v);
    }
}

// ------------------------------------------------------------- WMMA GEMM
// Y[R,NCOLS] = load(X)[R,K] @ W[K,NCOLS], optional column stats accumulation.
// W is staged in LDS pair-interleaved: element (k, c) lives at
//   sW[(k>>1)*2*NCOLS + 2*c + (k&1)]
// so each B fragment {W[kb][c], W[kb+1][c]} (kb even) is one aligned b64 load
// straight into the adjacent VGPR pair WMMA SRC1 wants (no v_mov shuffles).
enum { MODE_PLAIN = 0, MODE_ESUM = 1, MODE_EPROD = 2, MODE_CAT4 = 3 };

template <int MODE, int K, int NCOLS, bool STATS>
__global__ __launch_bounds__(TPB) void gemm_wmma(
    const float* __restrict__ X,
    const float* __restrict__ Eb, const float* __restrict__ Vb,
    const int* __restrict__ src, const int* __restrict__ dst,
    const float* __restrict__ P0, const float* __restrict__ P1,
    const float* __restrict__ P2, const float* __restrict__ P3,
    const float* __restrict__ W,
    float* __restrict__ Y, float* __restrict__ stats, int R) {
    constexpr int KC = 64;               // k-chunk staged in LDS
    constexpr int NKC = K / KC;
    constexpr int NT = NCOLS / 16;
    __shared__ float sW[KC * NCOLS];     // pair-interleaved layout (see above)

    const int lane = threadIdx.x & 31;
    const int wave = threadIdx.x >> 5;
    const int hh = lane >> 4;            // half-wave select (K pairs / M halves)
    const int l = lane & 15;
    const int ntiles = R >> 4;
    const int tile = blockIdx.x * WAVES + wave;
    const bool alive = tile < ntiles;    // wave-uniform: EXEC stays all-1 for WMMA
    const int row0 = tile << 4;
    const int myrow = row0 + l;

    int sidx = 0, didx = 0;
    if ((MODE == MODE_ESUM || MODE == MODE_EPROD) && alive) {
        sidx = src[myrow];
        didx = dst[myrow];
    }

    v8f c[NT];
#pragma unroll
    for (int nt = 0; nt < NT; ++nt)
#pragma unroll
        for (int i = 0; i < 8; ++i) c[nt][i] = 0.f;

    for (int kc = 0; kc < NKC; ++kc) {
        __syncthreads();
        for (int i = threadIdx.x; i < KC * NCOLS; i += TPB) {
            int kr = i / NCOLS, cc = i % NCOLS;
            sW[(kr >> 1) * (2 * NCOLS) + 2 * cc + (kr & 1)] =
                W[(size_t)(kc * KC + kr) * NCOLS + cc];
        }
        __syncthreads();

        if (alive) {
            // A fragments: 16x4 f32 layout — lanes 0-15 hold K0/K1, 16-31 K2/K3
            v2f a[16];
#pragma unroll
            for (int ks = 0; ks < 16; ++ks) {
                const int kl = ks * 4 + 2 * hh;   // within chunk
                float x0, x1;
                if (MODE == MODE_PLAIN) {
                    const float2 t = *(const float2*)(X + (size_t)myrow * K + kc * KC + kl);
                    x0 = t.x; x1 = t.y;
                } else if (MODE == MODE_ESUM) {
                    const float2 te = *(const float2*)(Eb + (size_t)myrow * D + kl);
                    const float2 ts = *(const float2*)(Vb + (size_t)sidx * D + kl);
                    const float2 td = *(const float2*)(Vb + (size_t)didx * D + kl);
                    x0 = te.x + ts.x + td.x; x1 = te.y + ts.y + td.y;
                } else if (MODE == MODE_EPROD) {
                    const float2 ts = *(const float2*)(Vb + (size_t)sidx * D + kl);
                    const float2 td = *(const float2*)(Vb + (size_t)didx * D + kl);
                    x0 = ts.x * td.x; x1 = ts.y * td.y;
                } else {  // MODE_CAT4: each 64-wide K block is one leaky() input tensor
                    const float* Pp = (kc == 0) ? P0 : (kc == 1) ? P1 : (kc == 2) ? P2 : P3;
                    const float2 t = *(const float2*)(Pp + (size_t)myrow * D + kl);
                    x0 = leaky_f(t.x); x1 = leaky_f(t.y);
                }
                a[ks].x = x0; a[ks].y = x1;
            }
#pragma unroll
            for (int nt = 0; nt < NT; ++nt) {
#pragma unroll
                for (int ks = 0; ks < 16; ++ks) {
                    const int pair = ks * 2 + hh;              // (ks*4 + 2*hh) >> 1
                    const v2f b = *(const v2f*)(sW + pair * (2 * NCOLS) + 2 * (nt * 16 + l));
                    c[nt] = __builtin_amdgcn_wmma_f32_16x16x4_f32(
                        false, a[ks], false, b, (short)0, c[nt], false, false);
                }
            }
        }
    }

    if (alive) {
#pragma unroll
        for (int nt = 0; nt < NT; ++nt) {
            float s = 0.f, sq = 0.f;
#pragma unroll
            for (int i = 0; i < 8; ++i) {
                float v = c[nt][i];
                // C layout: VGPR i -> row (row0 + 8*hh + i), col = nt*16 + l
                Y[(size_t)(row0 + 8 * hh + i) * NCOLS + nt * 16 + l] = v;
                s += v; sq += v * v;
            }
            if (STATS) {
                atomicAdd(stats + nt * 16 + l, s);
                atomicAdd(stats + NCOLS + nt * 16 + l, sq);
            }
        }
    }
}

// ------------------------------------------------------------------- driver
extern "C" void kernel_launch(void* const* d_in, const int* in_sizes, int n_in,
                              void* d_out, int out_size, void* d_ws, size_t ws_size,
                              hipStream_t stream) {
    (void)n_in; (void)out_size; (void)ws_size;
    const float* V0 = (const float*)d_in[0];
    const float* V1 = (const float*)d_in[1];
    const float* E0 = (const float*)d_in[2];
    const float* E1 = (const float*)d_in[3];
    const int*   src = (const int*)d_in[4];
    const int*   dst = (const int*)d_in[5];
    const float* arch = (const float*)d_in[6];
    const float* Wv = (const float*)d_in[7];
    const float* We = (const float*)d_in[8];
    const float* Wv1 = (const float*)d_in[9];
    const float* Wv2 = (const float*)d_in[10];
    const float* We1 = (const float*)d_in[11];
    const float* We2 = (const float*)d_in[12];

    const int N = in_sizes[0] / D;
    const int M = in_sizes[4];

    float* ws = (float*)d_ws;
    size_t off = 0;
    float* V2 = ws + off; off += (size_t)N * D;
    float* V3 = ws + off; off += (size_t)N * D;
    float* E2 = ws + off; off += (size_t)M * D;
    float* E3 = ws + off; off += (size_t)M * D;
    float* AGS = ws + off; off += (size_t)N * D;
    float* AGM = ws + off; off += (size_t)N * D;
    float* Yb = ws + off; off += (size_t)M * 128;   // shared GEMM output scratch
    float* ST = ws + off; off += 256;               // [colsum | colsumsq], <=128 cols
    float* Pb = ws + off; off += 32;                // softmaxed arch weights [8,3]

    float* outV = (float*)d_out;
    float* outE = outV + (size_t)N * D;

    auto fill = [&](float* p, size_t n, float v) {
        unsigned b = (unsigned)((n + 255) / 256);
        fill_f32<<<b, 256, 0, stream>>>(p, n, v);
    };
    auto bnacc = [&](const float* Y, int pidx, float* out, int R) {
        size_t n = (size_t)R * D;
        bn_accum<<<(unsigned)((n + 255) / 256), 256, 0, stream>>>(Y, ST, Pb, pidx, out, R, D);
    };

    softmax8x3<<<1, 8, 0, stream>>>(arch, Pb);

    const float NEG_INF = -__builtin_huge_valf();

    // -------- mixed_v: candidates {Vs, seg_sum(msg), seg_max(msg)} --------
    auto mixed_v = [&](const float* Vin, const float* Ein, int lv, float* OUT) {
        fill(AGS, (size_t)N * D, 0.f);
        fill(AGM, (size_t)N * D, NEG_INF);
        size_t mt = (size_t)M * 16;
        edge_msg_agg<<<(unsigned)((mt + 255) / 256), 256, 0, stream>>>(
            Vin, Ein, src, dst, AGS, AGM, M);
        fix_nonfinite<<<(unsigned)(((size_t)N * D + 255) / 256), 256, 0, stream>>>(
            AGM, (size_t)N * D);
        const float* cands[3] = {Vin, AGS, AGM};
        unsigned gb = (unsigned)((N / 16 + WAVES - 1) / WAVES);
        for (int i = 0; i < 3; ++i) {
            fill(ST, 128, 0.f);
            gemm_wmma<MODE_PLAIN, 64, 64, true><<<gb, TPB, 0, stream>>>(
                cands[i], nullptr, nullptr, nullptr, nullptr,
                nullptr, nullptr, nullptr, nullptr,
                Wv + (size_t)(lv * 3 + i) * D * D, Yb, ST, N);
            bnacc(Yb, lv * 3 + i, OUT, N);
        }
    };

    // -------- mixed_e: candidates {Es, vs+vd+Es, vs*vd} --------
    auto mixed_e = [&](const float* Vin, const float* Ein, int le, float* OUT) {
        unsigned gb = (unsigned)((M / 16 + WAVES - 1) / WAVES);
        for (int i = 0; i < 3; ++i) {
            fill(ST, 128, 0.f);
            const float* Wp = We + (size_t)(le * 3 + i) * D * D;
            if (i == 0) {
                gemm_wmma<MODE_PLAIN, 64, 64, true><<<gb, TPB, 0, stream>>>(
                    Ein, nullptr, nullptr, nullptr, nullptr,
                    nullptr, nullptr, nullptr, nullptr, Wp, Yb, ST, M);
            } else if (i == 1) {
                gemm_wmma<MODE_ESUM, 64, 64, true><<<gb, TPB, 0, stream>>>(
                    nullptr, Ein, Vin, src, dst,
                    nullptr, nullptr, nullptr, nullptr, Wp, Yb, ST, M);
            } else {
                gemm_wmma<MODE_EPROD, 64, 64, true><<<gb, TPB, 0, stream>>>(
                    nullptr, nullptr, Vin, src, dst,
                    nullptr, nullptr, nullptr, nullptr, Wp, Yb, ST, M);
            }
            bnacc(Yb, (4 + le) * 3 + i, OUT, M);
        }
    };

    // -------- DARTS cell (TOPO = (((0,0),(1,1)), ((1,1),(2,2)))) --------
    fill(V2, (size_t)N * D, 0.f);
    mixed_v(V0, E0, 0, V2);
    mixed_v(V1, E1, 1, V2);
    fill(E2, (size_t)M * D, 0.f);
    mixed_e(V0, E0, 0, E2);
    mixed_e(V1, E1, 1, E2);
    fill(V3, (size_t)N * D, 0.f);
    mixed_v(V1, E1, 2, V3);
    mixed_v(V2, E2, 3, V3);
    fill(E3, (size_t)M * D, 0.f);
    mixed_e(V1, E1, 2, E3);
    mixed_e(V2, E2, 3, E3);

    // -------- final heads: leaky(BN(cat4 @ W1)) @ W2 --------
    {   // V head
        fill(ST, 256, 0.f);
        unsigned gb = (unsigned)((N / 16 + WAVES - 1) / WAVES);
        gemm_wmma<MODE_CAT4, 256, 128, true><<<gb, TPB, 0, stream>>>(
            nullptr, nullptr, nullptr, nullptr, nullptr,
            V0, V1, V2, V3, Wv1, Yb, ST, N);
        size_t n = (size_t)N * 128;
        bn_leaky_inplace<<<(unsigned)((n + 255) / 256), 256, 0, stream>>>(Yb, ST, N, 128);
        gemm_wmma<MODE_PLAIN, 128, 64, false><<<gb, TPB, 0, stream>>>(
            Yb, nullptr, nullptr, nullptr, nullptr,
            nullptr, nullptr, nullptr, nullptr, Wv2, outV, ST, N);
    }
    {   // E head
        fill(ST, 256, 0.f);
        unsigned gb = (unsigned)((M / 16 + WAVES - 1) / WAVES);
        gemm_wmma<MODE_CAT4, 256, 128, true><<<gb, TPB, 0, stream>>>(
            nullptr, nullptr, nullptr, nullptr, nullptr,
            E0, E1, E2, E3, We1, Yb, ST, M);
        size_t n = (size_t)M * 128;
        bn_leaky_inplace<<<(unsigned)((n + 255) / 256), 256, 0, stream>>>(Yb, ST, M, 128);
        gemm_wmma<MODE_PLAIN, 128, 64, false><<<gb, TPB, 0, stream>>>(
            Yb, nullptr, nullptr, nullptr, nullptr,
            nullptr, nullptr, nullptr, nullptr, We2, outE, ST, M);
    }
}